// S5ContrastiveEncoder_70832600646410
// MI455X (gfx1250) — compile-verified
//
#include <hip/hip_runtime.h>
#include <hip/hip_bf16.h>

typedef __attribute__((ext_vector_type(2))) float v2f;
typedef __attribute__((ext_vector_type(8))) float v8f;

#define B_      32
#define S_      4096
#define D_      64
#define P_      8
#define STRIDE_ 4
#define H_      512
#define T_      1023          // (4096-8)/4+1
#define KTOT    512           // P*D
#define EPS_    1e-6f

#define M_TOT   (B_*T_)       // 32736 = 1023 blocks * 32
#define MBLK    32
#define KC      16
#define A_PAD   48            // (48*2) % 64 == 32 -> disjoint bank halves
#define B_PAD   528           // (528*2) % 64 == 32 -> disjoint bank halves

// d_out layout (float elements): hidden | token_lengths | tokens
#define OFF_LEN  ((size_t)B_*T_*H_)
#define OFF_TOK  (OFF_LEN + B_)

__device__ __forceinline__ int ntok_of(int len) {
    return len >= P_ ? (len - P_) / STRIDE_ + 1 : 1;
}

// ---------------- tokens materialization (bandwidth-bound) ----------------
__global__ __launch_bounds__(256) void tokens_kernel(
    const float* __restrict__ x, const int* __restrict__ lengths,
    float* __restrict__ out)
{
    int f = blockIdx.x * blockDim.x + threadIdx.x;   // float4 index
    const int NF4 = B_ * T_ * (KTOT / 4);
    if (f >= NF4) return;
    int row = f >> 7;            // tokens row (512 floats = 128 f4)
    int c4  = f & 127;
    int k   = c4 << 2;
    int p   = k >> 6;
    int d   = k & 63;
    int b   = row / T_;
    int t   = row - b * T_;
    int len = lengths[b];
    int pos = t * STRIDE_ + p;
    float4 v = make_float4(0.f, 0.f, 0.f, 0.f);
    if (pos < len && t < ntok_of(len))
        v = *(const float4*)(x + ((size_t)b * S_ + pos) * D_ + d);
    *(float4*)(out + OFF_TOK + (size_t)row * KTOT + k) = v;
}

__global__ void lengths_kernel(const int* __restrict__ lengths, float* __restrict__ out)
{
    int i = threadIdx.x;
    if (i < B_) out[OFF_LEN + i] = (float)ntok_of(lengths[i]);
}

// ---------------- fused gather + GEMM + RMS-norm (WMMA f32) ----------------
// K-pairs interleaved in LDS so every fragment is one aligned ds_load_b64.
__global__ __launch_bounds__(256) void gemm_rms_kernel(
    const float* __restrict__ x, const int* __restrict__ lengths,
    const float* __restrict__ W, const float* __restrict__ bias,
    const float* __restrict__ scale, float* __restrict__ out)
{
    __shared__ float As2[KC/2][A_PAD][2];   // {A[m][2r], A[m][2r+1]}
    __shared__ float Bs2[KC/2][B_PAD][2];   // {W[2r][n], W[2r+1][n]}
    __shared__ float row_ssq[MBLK];

    const int tid    = threadIdx.x;
    const int wid    = tid >> 5;
    const int lane   = tid & 31;
    const int lh     = lane & 15;     // lane within half
    const int hsel   = lane >> 4;     // 0: lanes 0-15, 1: lanes 16-31
    const int wave_m = wid >> 2;      // 0..1  (16-row strip)
    const int wave_n = wid & 3;       // 0..3  (128-col group)
    const int m0     = blockIdx.x * MBLK;

    if (tid < MBLK) row_ssq[tid] = 0.0f;

    v8f acc[8];
#pragma unroll
    for (int nt = 0; nt < 8; ++nt) { v8f z = {}; acc[nt] = z; }

    // A staging coords: each thread gathers one K-pair (2 consecutive k)
    const int a_m  = (tid * 2) / KC;        // 0..31
    const int a_kk = (tid * 2) % KC;        // even
    const int gm   = m0 + a_m;
    const int ab   = gm / T_;
    const int at   = gm - ab * T_;
    const int alen = lengths[ab];
    const bool arow_live = (at < ntok_of(alen));

    // B staging coords precompute: u = tid + i*256; rp = u>>8; n = (u&255)*2
    for (int kc = 0; kc < KTOT / KC; ++kc) {
        // stage A tile (32 x 16): masked gather straight from x (L2 resident)
        {
            int kbase = kc * KC + a_kk;     // even -> d,d+1 in same patch row
            int p   = kbase >> 6;
            int d   = kbase & 63;
            int pos = at * STRIDE_ + p;
            float2 v = make_float2(0.f, 0.f);
            if (arow_live && pos < alen)
                v = *(const float2*)(x + ((size_t)ab * S_ + pos) * D_ + d);
            *(float2*)(&As2[a_kk >> 1][a_m][0]) = v;
        }
        // stage B tile (16 x 512), K-pair interleaved: 8 float4 stores/thread
        {
            const float* wsrc = W + (size_t)kc * KC * H_;
#pragma unroll
            for (int i = 0; i < 8; ++i) {
                int u  = tid + i * 256;
                int rp = u >> 8;               // row pair 0..7
                int n  = (u & 255) << 1;       // even column
                const float* w0 = wsrc + (size_t)(2 * rp) * H_ + n;
                float2 r0 = *(const float2*)(w0);        // W[2rp][n..n+1]
                float2 r1 = *(const float2*)(w0 + H_);   // W[2rp+1][n..n+1]
                float4 pk = make_float4(r0.x, r1.x, r0.y, r1.y);
                *(float4*)(&Bs2[rp][n][0]) = pk;
            }
        }
        __syncthreads();

#pragma unroll
        for (int ks = 0; ks < KC; ks += 4) {
            const int rp = (ks >> 1) + hsel;  // ISA 16x4 f32 frag layout
            v2f a = *(const v2f*)(&As2[rp][wave_m * 16 + lh][0]);
#pragma unroll
            for (int nt = 0; nt < 8; ++nt) {
                const int col = wave_n * 128 + nt * 16 + lh;
                v2f bf = *(const v2f*)(&Bs2[rp][col][0]);
                acc[nt] = __builtin_amdgcn_wmma_f32_16x16x4_f32(
                    false, a, false, bf, (short)0, acc[nt], false, false);
            }
        }
        __syncthreads();
    }

    // ---------------- epilogue: bias + RMS-norm + scale ----------------
    float bcol[8], scol[8];
#pragma unroll
    for (int nt = 0; nt < 8; ++nt) {
        const int col = wave_n * 128 + nt * 16 + lh;
        bcol[nt] = bias[col];
        scol[nt] = scale[col];
    }
    float ssq[8];
#pragma unroll
    for (int r = 0; r < 8; ++r) ssq[r] = 0.f;
#pragma unroll
    for (int nt = 0; nt < 8; ++nt)
#pragma unroll
        for (int r = 0; r < 8; ++r) {
            float h = acc[nt][r] + bcol[nt];
            acc[nt][r] = h;
            ssq[r] += h * h;
        }
    // reduce across the 16 lanes of each half-wave (masks < 16 stay in-half)
#pragma unroll
    for (int r = 0; r < 8; ++r) {
        float s = ssq[r];
        s += __shfl_xor(s, 1, 32);
        s += __shfl_xor(s, 2, 32);
        s += __shfl_xor(s, 4, 32);
        s += __shfl_xor(s, 8, 32);
        ssq[r] = s;
    }
    if (lh == 0) {
#pragma unroll
        for (int r = 0; r < 8; ++r)
            atomicAdd(&row_ssq[wave_m * 16 + hsel * 8 + r], ssq[r]); // ds_add_f32
    }
    __syncthreads();

    float inv_rms[8];
#pragma unroll
    for (int r = 0; r < 8; ++r) {
        float s = row_ssq[wave_m * 16 + hsel * 8 + r];
        inv_rms[r] = rsqrtf(s * (1.0f / H_) + EPS_);
    }
#pragma unroll
    for (int nt = 0; nt < 8; ++nt) {
        const int col = wave_n * 128 + nt * 16 + lh;
#pragma unroll
        for (int r = 0; r < 8; ++r) {
            const int rowg = m0 + wave_m * 16 + hsel * 8 + r;
            out[(size_t)rowg * H_ + col] = acc[nt][r] * inv_rms[r] * scol[nt];
        }
    }
}

extern "C" void kernel_launch(void* const* d_in, const int* in_sizes, int n_in,
                              void* d_out, int out_size, void* d_ws, size_t ws_size,
                              hipStream_t stream) {
    (void)in_sizes; (void)n_in; (void)out_size; (void)d_ws; (void)ws_size;
    const float* x       = (const float*)d_in[0];
    const int*   lengths = (const int*)d_in[1];
    const float* W       = (const float*)d_in[2];
    const float* bias    = (const float*)d_in[3];
    const float* scale   = (const float*)d_in[4];
    float* out = (float*)d_out;

    lengths_kernel<<<1, 32, 0, stream>>>(lengths, out);

    const int NF4 = B_ * T_ * (KTOT / 4);
    tokens_kernel<<<(NF4 + 255) / 256, 256, 0, stream>>>(x, lengths, out);

    gemm_rms_kernel<<<M_TOT / MBLK, 256, 0, stream>>>(x, lengths, W, bias, scale, out);
}